// DifferentiableVoxelRasterizer_17514876634266
// MI455X (gfx1250) — compile-verified
//
#include <hip/hip_runtime.h>
#include <hip/hip_bf16.h>
#include <math.h>

// ---------------------------------------------------------------------------
// DifferentiableVoxelRasterizer for gfx1250 (MI455X, wave32)
//
// Kernel 1 (1 block / 512 threads = 16 waves):
//   Phase A: cam = hom @ camera_matrix.T via V_WMMA_F32_16X16X4_F32
//            (32 tiles of 16 voxel rows; B = M^T zero-padded to 4x16,
//             hoisted out of the tile loop since it is loop-invariant)
//   Phase B: per-voxel projection, visibility, alpha (softplus), LDS SoA
//   Phase C: 512-wide bitonic sort in LDS, key = depth descending,
//            tie-break = index ascending (matches stable argsort(-depth))
//   Phase D: gather sorted SoA -> workspace (8 x 512 floats = 16 KB)
//
// Kernel 2 (256 blocks x 256 threads; one image row per block):
//   Stage 16 KB sorted SoA into LDS via GLOBAL_LOAD_ASYNC_TO_LDS_B128
//   (ASYNCcnt path, bypasses VGPRs), then each pixel runs the 512-step
//   blend recurrence fully in registers; single framebuffer write.
// ---------------------------------------------------------------------------

#define NVOX   512
#define IMG_W  256
#define IMG_H  256
#define FARV   100.0f
#define NEARV  0.1f

typedef __attribute__((ext_vector_type(2))) float v2f;
typedef __attribute__((ext_vector_type(8))) float v8f;
typedef __attribute__((ext_vector_type(4))) int   v4i;

__global__ __launch_bounds__(512)
void voxrast_project_sort(const float* __restrict__ pos,     // [512,3]
                          const float* __restrict__ sizes,   // [512]
                          const float* __restrict__ dens,    // [512]
                          const float* __restrict__ cols,    // [512,3]
                          const float* __restrict__ cm,      // [4,4]
                          const float* __restrict__ intr,    // [3,3]
                          float* __restrict__ ws)            // [8,512]
{
    __shared__ float s_cam[NVOX * 4];
    __shared__ float s_sx[NVOX], s_sy[NVOX], s_hs[NVOX], s_va[NVOX];
    __shared__ float s_dep[NVOX], s_cr[NVOX], s_cg[NVOX], s_cb[NVOX];
    __shared__ float s_key[NVOX];
    __shared__ int   s_idx[NVOX];

    const int tid  = threadIdx.x;
    const int lane = tid & 31;
    const int wave = tid >> 5;
    const int half = lane >> 4;     // 0: lanes 0-15, 1: lanes 16-31
    const int l15  = lane & 15;

    // ---------------- Phase A: cam = hom @ M^T via WMMA f32 16x16x4 -------
    // A (16x4 f32): lanes 0-15 hold K=0,1 ; lanes 16-31 hold K=2,3
    // B (4x16 f32): lanes 0-15 hold K=0,1 ; lanes 16-31 hold K=2,3
    //   B[k][n] = M^T[k][n] = M[n][k], zero for n >= 4
    v2f b;                                  // loop-invariant: load once
    if (l15 < 4) {
        b.x = cm[l15 * 4 + 0 + 2 * half];
        b.y = cm[l15 * 4 + 1 + 2 * half];
    } else {
        b.x = 0.0f; b.y = 0.0f;
    }

    for (int t = wave; t < NVOX / 16; t += 16) {   // 32 tiles over 16 waves
        const int v0 = t * 16;
        const int m  = v0 + l15;

        v2f a;
        a.x = pos[m * 3 + 2 * half];          // half0: x, half1: z
        a.y = half ? 1.0f : pos[m * 3 + 1];   // half0: y, half1: 1

        v8f c = {};
        // 8 args: (neg_a, A, neg_b, B, c_mod, C, reuse_a, reuse_b)
        v8f d = __builtin_amdgcn_wmma_f32_16x16x4_f32(
                    false, a, false, b, (short)0, c, false, false);

        // D layout: VGPR r, lane L -> row = r + 8*(L>=16), col = L&15
        if (l15 < 4) {
            const int base = v0 + half * 8;
#pragma unroll
            for (int r = 0; r < 8; ++r)
                s_cam[(base + r) * 4 + l15] = d[r];
        }
    }
    __syncthreads();

    // ---------------- Phase B: per-voxel projection -----------------------
    {
        const int v = tid;   // 512 threads, one voxel each
        const float c0 = s_cam[v * 4 + 0];
        const float c1 = s_cam[v * 4 + 1];
        const float c2 = s_cam[v * 4 + 2];
        const float c3 = s_cam[v * 4 + 3];
        const float invw = 1.0f / c3;
        const float x3 = c0 * invw, y3 = c1 * invw, z3 = c2 * invw;

        const float K00 = intr[0], K01 = intr[1], K02 = intr[2];
        const float K10 = intr[3], K11 = intr[4], K12 = intr[5];
        const float K20 = intr[6], K21 = intr[7], K22 = intr[8];

        const float srx = K00 * x3 + K01 * y3 + K02 * z3;
        const float sry = K10 * x3 + K11 * y3 + K12 * z3;
        const float srz = K20 * x3 + K21 * y3 + K22 * z3;
        const float invz = 1.0f / srz;
        const float px = srx * invz;
        const float py = sry * invz;
        const float dep = z3;

        const float size = sizes[v];
        const float ss = size * K00 / fmaxf(dep, 0.1f);

        const bool vis = (dep > NEARV) && (dep < FARV) &&
                         (px + ss >= 0.0f) && (px - ss < (float)IMG_W) &&
                         (py + ss >= 0.0f) && (py - ss < (float)IMG_H);

        const float dn = dens[v];
        const float splus = (dn > 20.0f) ? dn : log1pf(expf(dn)); // softplus
        float va = 1.0f - expf(-splus * size);
        va = fminf(fmaxf(va, 0.0f), 1.0f);
        va = vis ? va : 0.0f;

        s_sx[v]  = px;
        s_sy[v]  = py;
        s_hs[v]  = 0.5f * ss;
        s_va[v]  = va;
        s_dep[v] = dep;
        s_cr[v]  = cols[v * 3 + 0];
        s_cg[v]  = cols[v * 3 + 1];
        s_cb[v]  = cols[v * 3 + 2];
        s_key[v] = dep;
        s_idx[v] = v;
    }

    // ---------------- Phase C: bitonic sort (depth desc, idx asc) ---------
    for (int k = 2; k <= NVOX; k <<= 1) {
        for (int j = k >> 1; j > 0; j >>= 1) {
            __syncthreads();
            const int i   = tid;
            const int ixj = i ^ j;
            if (ixj > i) {
                const bool dirDesc = ((i & k) == 0);
                const float ka = s_key[i],  kb = s_key[ixj];
                const int   ia = s_idx[i],  ib = s_idx[ixj];
                // strict total order: a precedes b iff deeper, tie -> lower idx
                const bool aPrecedesB = (ka > kb) || ((ka == kb) && (ia < ib));
                const bool doSwap = dirDesc ? (!aPrecedesB) : aPrecedesB;
                if (doSwap) {
                    s_key[i] = kb; s_key[ixj] = ka;
                    s_idx[i] = ib; s_idx[ixj] = ia;
                }
            }
        }
    }
    __syncthreads();

    // ---------------- Phase D: gather sorted SoA to workspace -------------
    {
        const int p = s_idx[tid];
        ws[0 * NVOX + tid] = s_sx[p];
        ws[1 * NVOX + tid] = s_sy[p];
        ws[2 * NVOX + tid] = s_hs[p];
        ws[3 * NVOX + tid] = s_va[p];
        ws[4 * NVOX + tid] = s_dep[p];
        ws[5 * NVOX + tid] = s_cr[p];
        ws[6 * NVOX + tid] = s_cg[p];
        ws[7 * NVOX + tid] = s_cb[p];
    }
}

__global__ __launch_bounds__(256)
void voxrast_blend(const float* __restrict__ ws,   // [8,512] sorted SoA
                   float* __restrict__ out)        // rgb | depth | alpha
{
    __shared__ __align__(16) float s[8 * NVOX];   // 16 KB voxel SoA

    // ---- Stage SoA into LDS -------------------------------------------
#if __has_builtin(__builtin_amdgcn_global_load_async_to_lds_b128)
    // CDNA5 async copy: memory -> LDS directly, tracked with ASYNCcnt.
    // Builtin signature (per toolchain diagnostic): v4i-vector pointers,
    // global source first, LDS destination second, imm offset, imm cpol.
    {
        typedef __attribute__((address_space(1))) v4i gv4i;
        typedef __attribute__((address_space(3))) v4i lv4i;
#pragma unroll
        for (int r = 0; r < 4; ++r) {
            const int i = threadIdx.x + r * 256;              // float4 index
            __builtin_amdgcn_global_load_async_to_lds_b128(
                (gv4i*)(ws + i * 4), (lv4i*)(s + i * 4), 0, 0);
        }
    }
#if __has_builtin(__builtin_amdgcn_s_wait_asynccnt)
    __builtin_amdgcn_s_wait_asynccnt(0);
#else
    asm volatile("s_wait_asynccnt 0x0" ::: "memory");
#endif
#else
    // Fallback: 128-bit loads through VGPRs.
    {
        const float4* src = (const float4*)ws;
        float4*       dst = (float4*)s;
        for (int i = threadIdx.x; i < (8 * NVOX) / 4; i += 256)
            dst[i] = src[i];
    }
#endif
    __syncthreads();

    const float* vx = s + 0 * NVOX;
    const float* vy = s + 1 * NVOX;
    const float* vh = s + 2 * NVOX;
    const float* va = s + 3 * NVOX;
    const float* vd = s + 4 * NVOX;
    const float* vr = s + 5 * NVOX;
    const float* vg = s + 6 * NVOX;
    const float* vb = s + 7 * NVOX;

    const float px = (float)threadIdx.x;
    const float py = (float)blockIdx.x;

    float cr = 0.0f, cg = 0.0f, cb = 0.0f;   // BG = 0
    float db = FARV;
    float ab = 0.0f;

    for (int v = 0; v < NVOX; ++v) {
        const float dx = px - vx[v];
        const float dy = py - vy[v];
        const float d2 = dx * dx + dy * dy;
        const float dist = sqrtf(fmaxf(d2, 1e-12f));
        float w = expf(-0.5f * dist);          // SIGMA = 1 -> exp(-dist/2) in [0,1]
        w = (dist <= vh[v]) ? w : 0.0f;
        float blend = va[v] * w * (1.0f - ab);
        blend = fminf(fmaxf(blend, 0.0f), 1.0f);
        const float om = 1.0f - blend;
        cr = cr * om + vr[v] * blend;
        cg = cg * om + vg[v] * blend;
        cb = cb * om + vb[v] * blend;
        db = db * om + vd[v] * blend;
        ab = ab + blend;
    }

    const int pix = (int)blockIdx.x * IMG_W + (int)threadIdx.x;
    out[pix * 3 + 0] = cr;
    out[pix * 3 + 1] = cg;
    out[pix * 3 + 2] = cb;
    out[IMG_H * IMG_W * 3 + pix] = db;                       // depth plane
    out[IMG_H * IMG_W * 3 + IMG_H * IMG_W + pix] = ab;       // alpha plane
}

extern "C" void kernel_launch(void* const* d_in, const int* in_sizes, int n_in,
                              void* d_out, int out_size, void* d_ws, size_t ws_size,
                              hipStream_t stream) {
    const float* pos   = (const float*)d_in[0];   // positions [512,3]
    const float* sizes = (const float*)d_in[1];   // sizes     [512]
    const float* dens  = (const float*)d_in[2];   // densities [512]
    const float* cols  = (const float*)d_in[3];   // colors    [512,3]
    const float* cm    = (const float*)d_in[4];   // camera    [4,4]
    const float* intr  = (const float*)d_in[5];   // intrinsics[3,3]

    float* ws  = (float*)d_ws;      // needs 8*512*4 = 16 KB
    float* out = (float*)d_out;     // 327680 floats: rgb | depth | alpha

    voxrast_project_sort<<<1, 512, 0, stream>>>(pos, sizes, dens, cols, cm, intr, ws);
    voxrast_blend<<<IMG_H, IMG_W, 0, stream>>>(ws, out);
}